// DynamicGraphEmbedding_16827681866102
// MI455X (gfx1250) — compile-verified
//
#include <hip/hip_runtime.h>
#include <hip/hip_bf16.h>
#include <math.h>
#include <stdint.h>

// Problem constants (match reference)
#define NN 256   // nodes
#define TT 256   // feature dim
#define BB 64    // batch
#define DD 64    // embedding dim
#define KK 16    // top-k

typedef __attribute__((ext_vector_type(2))) float        v2f;
typedef __attribute__((ext_vector_type(8))) float        v8f;
typedef __attribute__((ext_vector_type(4))) unsigned int v4u;
typedef __attribute__((ext_vector_type(4))) int          v4i;
typedef __attribute__((ext_vector_type(8))) int          v8i;

#if defined(__has_builtin)
# if __has_builtin(__builtin_amdgcn_tensor_load_to_lds)
#  define CDNA5_HAS_TDM 1
# endif
#endif
#ifndef CDNA5_HAS_TDM
# define CDNA5_HAS_TDM 0
#endif

// ---------------------------------------------------------------------------
// K1: build graph. One block per destination node i.
// ---------------------------------------------------------------------------
__global__ void dge_build_graph(const float* __restrict__ emb,
                                const float* __restrict__ logits,
                                const float* __restrict__ gumbel_u,
                                int* __restrict__ srcIdx,
                                float* __restrict__ wgt) {
    const int i = blockIdx.x;
    const int j = threadIdx.x;

    __shared__ float embi[DD];
    __shared__ float cosv[NN];
    __shared__ float rv[NN];
    __shared__ int   ri[NN];

    if (j < DD) embi[j] = emb[i * DD + j];
    __syncthreads();

    float dot = 0.f, nj = 0.f, ni = 0.f;
#pragma unroll
    for (int d = 0; d < DD; ++d) {
        float ej = emb[j * DD + d];
        float ei = embi[d];
        dot += ei * ej;
        nj  += ej * ej;
        ni  += ei * ei;
    }
    cosv[j] = dot * __frsqrt_rn(ni * nj);
    __syncthreads();

    for (int it = 0; it < KK; ++it) {
        rv[j] = cosv[j];
        ri[j] = j;
        __syncthreads();
#pragma unroll
        for (int s = NN / 2; s > 0; s >>= 1) {
            if (j < s) {
                float ov = rv[j + s];
                int   oi = ri[j + s];
                if (ov > rv[j] || (ov == rv[j] && oi < ri[j])) {
                    rv[j] = ov;
                    ri[j] = oi;
                }
            }
            __syncthreads();
        }
        if (j == 0) {
            int jm = ri[0];
            int e  = i * NN + jm;                 // edge index dst*N + src
            float u0 = gumbel_u[e * 2 + 0];
            float u1 = gumbel_u[e * 2 + 1];
            float g0 = -__logf(-__logf(u0));
            float g1 = -__logf(-__logf(u1));
            float l0 = logits[e * 2 + 0];
            float l1 = logits[e * 2 + 1];
            float gate = (l0 + g0 >= l1 + g1) ? 1.0f : 0.0f;
            srcIdx[i * KK + it] = jm;
            wgt[i * KK + it]    = gate * 0.0625f; // ew = dinv*dinv = 1/16
            cosv[jm] = -INFINITY;
        }
        __syncthreads();
    }
}

// ---------------------------------------------------------------------------
// K2: y[b*N+i, :] = sum_k w[i,k] * x_flat[b*N + src[i,k], :]
// ---------------------------------------------------------------------------
__global__ void dge_aggregate(const float* __restrict__ x,
                              const int* __restrict__ srcIdx,
                              const float* __restrict__ wgt,
                              float* __restrict__ y) {
    const int r    = blockIdx.x;       // b*N + i
    const int node = r & (NN - 1);
    const int base = r - node;
    const int t    = threadIdx.x;

    float acc = 0.f;
#pragma unroll
    for (int k = 0; k < KK; ++k) {
        float wk = wgt[node * KK + k];     // uniform
        if (wk != 0.f) {                   // uniform branch
            int s = srcIdx[node * KK + k];
            acc += wk * x[(base + s) * TT + t];
        }
    }
    y[r * TT + t] = acc;
}

// ---------------------------------------------------------------------------
// K3: out = y @ W + bias  via V_WMMA_F32_16X16X4_F32.
//   Block: 256 thr / 8 waves; tile 128(M) x 64(N); wave 32x32 (2x2 accs).
//
//   B path: W[:, n0:n0+64] transposed into LDS (sWT[col][k], stride 260 dw)
//     -> B fragment {W[k][n], W[k+1][n]} is contiguous => one ds_load_b64.
//        Banks (4c+k)%64: halves (k vs k+2) hit disjoint sets, conflict-free.
//   A path: Tensor Data Mover double-buffer. Chunk = 128 rows x 64 k, TDM
//     LDS padding (interval=64dw code 5, amount=4dw code 3) => row stride 68
//     -> A fragment is one ds_load_b64, same disjoint-bank-halves property.
//     Wave0: issue chunk c+1, s_wait_tensorcnt 1 (TDM is in-order => chunk c
//     done), barrier, all waves compute, barrier (frees buffer c&1 before
//     chunk c+2 overwrites it).
//   Steady state inner loop: 4x ds_load_b64 + 4x v_wmma, zero vmem.
//   LDS: 2*128*68*4 + 64*260*4 = 136,192 B => 2 blocks per 320KB WGP.
// ---------------------------------------------------------------------------
#define AS   68            // A chunk row stride (dwords)
#define BS   260           // W^T column stride (dwords)
#define ACH  (128 * AS)    // A chunk size (dwords)

__device__ __forceinline__ v8f wmma_f32_4(v2f a, v2f b, v8f c) {
    return __builtin_amdgcn_wmma_f32_16x16x4_f32(
        false, a, false, b, (short)0, c, false, false);
}

#if CDNA5_HAS_TDM
// TDM: load 128 rows x 64 cols of row-major [*,256] f32 tensor into LDS with
// 4-dword padding per 64-dword row (D# per ISA 8.3/8.4).
__device__ __forceinline__ void tdm_load_tile(const float* gsrc, unsigned lds_off) {
    unsigned long long ga = (unsigned long long)(uintptr_t)gsrc;

    v4u g0;                                       // D# group 0
    g0.x = 1u;                                    // count=1, user, no gather
    g0.y = lds_off;                               // lds_addr (bytes)
    g0.z = (unsigned)(ga & 0xFFFFFFFFull);        // global_addr[31:0]
    g0.w = (unsigned)((ga >> 32) & 0x01FFFFFFull) // global_addr[56:32]
         | (2u << 30);                            // type = 2

    v8i g1;                                       // D# group 1
    g1[0] = (int)0x07520000u;   // data_size=4B(2), pad_en, intvl=64dw(5), amt=4dw(3)
    g1[1] = (int)(256u   << 16);  // tensor_dim0 = 256
    g1[2] = (int)(16384u << 16);  // tensor_dim1 = 16384 (rows, OOB bound)
    g1[3] = (int)(64u    << 16);  // tensor_dim1 hi=0 | tile_dim0 = 64
    g1[4] = 128;                  // tile_dim1 = 128 rows
    g1[5] = 256;                  // tensor_dim0_stride = 256 elements
    g1[6] = 0;                    // stride0 hi / stride1 lo
    g1[7] = 1;                    // tensor_dim1_stride = 65536 (bits 47:16)

    v4i z4 = {};
#if __clang_major__ >= 23
    v8i z8 = {};
    __builtin_amdgcn_tensor_load_to_lds(g0, g1, z4, z4, z8, 0);
#else
    __builtin_amdgcn_tensor_load_to_lds(g0, g1, z4, z4, 0);
#endif
}
#endif

__global__ void dge_gemm_wmma(const float* __restrict__ Y,
                              const float* __restrict__ W,
                              const float* __restrict__ bias,
                              float* __restrict__ out) {
    __shared__ __align__(16) float sA[2 * ACH];   // 69,632 B (TDM double buffer)
    __shared__ __align__(16) float sWT[64 * BS];  // 66,560 B (W slice, transposed)

    const int lane   = threadIdx.x & 31;
    const int waveId = threadIdx.x >> 5;
    const int wm     = waveId >> 1;    // 0..3
    const int wn     = waveId & 1;     // 0..1

    const int m0blk = blockIdx.x * 128;
    const int n0blk = blockIdx.y * 64;
    const int m0    = m0blk + wm * 32;
    const int n0    = n0blk + wn * 32;

    const int half = lane >> 4;   // k-pair selector
    const int t    = lane & 15;   // row (A) / col (B,C,D) within tile

#if CDNA5_HAS_TDM
    // Kick A chunk 0 while the block transposes the W slice.
    if (waveId == 0)
        tdm_load_tile(Y + m0blk * TT, (unsigned)(uintptr_t)&sA[0]);
#endif

    // Cooperative transpose: sWT[c*BS + k] = W[k][n0blk + c].
    // Reads: 64-float contiguous row segments (coalesced).
    for (int idx = threadIdx.x; idx < TT * 64; idx += 256) {
        int kk = idx >> 6, c = idx & 63;
        sWT[c * BS + kk] = W[kk * TT + n0blk + c];
    }

    v8f c00 = {}, c01 = {}, c10 = {}, c11 = {};

    const float* pB = sWT + (wn * 32 + t) * BS + 2 * half;   // + k
    const float* pArow;

    for (int ch = 0; ch < 4; ++ch) {
#if CDNA5_HAS_TDM
        if (waveId == 0) {
            if (ch < 3) {
                tdm_load_tile(Y + m0blk * TT + (ch + 1) * 64,
                              (unsigned)(uintptr_t)&sA[((ch + 1) & 1) * ACH]);
                __builtin_amdgcn_s_wait_tensorcnt(1);  // chunk ch complete
            } else {
                __builtin_amdgcn_s_wait_tensorcnt(0);
            }
        }
#else
        // Fallback: cooperative copy of chunk ch into the same padded layout.
        for (int idx = threadIdx.x; idx < 128 * 64; idx += 256) {
            int rr = idx >> 6, kk = idx & 63;
            sA[(ch & 1) * ACH + rr * AS + kk] = Y[(m0blk + rr) * TT + ch * 64 + kk];
        }
#endif
        __syncthreads();   // chunk ch visible (and, on ch==0, sWT complete)

        pArow = sA + (ch & 1) * ACH + (wm * 32 + t) * AS + 2 * half;
        const float* pBc = pB + ch * 64;

#pragma unroll 8
        for (int kc = 0; kc < 64; kc += 4) {
            v2f a0 = *(const v2f*)(pArow + kc);            // rows m0+t
            v2f a1 = *(const v2f*)(pArow + 16 * AS + kc);  // rows m0+16+t
            v2f b0 = *(const v2f*)(pBc + kc);              // cols n0+t
            v2f b1 = *(const v2f*)(pBc + 16 * BS + kc);    // cols n0+16+t

            c00 = wmma_f32_4(a0, b0, c00);
            c01 = wmma_f32_4(a0, b1, c01);
            c10 = wmma_f32_4(a1, b0, c10);
            c11 = wmma_f32_4(a1, b1, c11);
        }
        __syncthreads();   // all waves done with buffer (ch&1) before reuse
    }

    // ---- epilogue: fused bias add, stores per ISA C/D layout --------------
    const float bz0 = bias[n0 + t];
    const float bz1 = bias[n0 + 16 + t];
#pragma unroll
    for (int v = 0; v < 8; ++v) {
        int mr = v + 8 * half;
        out[(m0 + mr)      * TT + n0 + t]      = c00[v] + bz0;
        out[(m0 + mr)      * TT + n0 + 16 + t] = c01[v] + bz1;
        out[(m0 + 16 + mr) * TT + n0 + t]      = c10[v] + bz0;
        out[(m0 + 16 + mr) * TT + n0 + 16 + t] = c11[v] + bz1;
    }
}

// ---------------------------------------------------------------------------
// Host launcher
// Inputs (setup_inputs order): 0:x [B,N,T] 1:emb [N,D] 2:W [T,T] 3:b [T]
//                              4:logits [N*N,2] 5:gumbel_u [N*N,2]
// ws layout: [0, 16KB)   int   src[N*K]
//            [16KB,32KB) float w[N*K]
//            [32KB, ...) float y[B*N*T]   (16.78 MB)
// ---------------------------------------------------------------------------
extern "C" void kernel_launch(void* const* d_in, const int* in_sizes, int n_in,
                              void* d_out, int out_size, void* d_ws, size_t ws_size,
                              hipStream_t stream) {
    (void)in_sizes; (void)n_in; (void)out_size; (void)ws_size;

    const float* x        = (const float*)d_in[0];
    const float* emb      = (const float*)d_in[1];
    const float* W        = (const float*)d_in[2];
    const float* bias     = (const float*)d_in[3];
    const float* logits   = (const float*)d_in[4];
    const float* gumbel_u = (const float*)d_in[5];
    float* out = (float*)d_out;

    char* ws = (char*)d_ws;
    int*   srcIdx = (int*)ws;                        // 4096 ints
    float* wgt    = (float*)(ws + 16 * 1024);        // 4096 floats
    float* y      = (float*)(ws + 32 * 1024);        // B*N*T floats

    dge_build_graph<<<NN, NN, 0, stream>>>(emb, logits, gumbel_u, srcIdx, wgt);
    dge_aggregate<<<BB * NN, TT, 0, stream>>>(x, srcIdx, wgt, y);

    dim3 g3((BB * NN) / 128, TT / 64);
    dge_gemm_wmma<<<g3, 256, 0, stream>>>(y, W, bias, out);
}